// GraphEncoder_43671227466074
// MI455X (gfx1250) — compile-verified
//
#include <hip/hip_runtime.h>

typedef float v2f __attribute__((ext_vector_type(2)));
typedef float v8f __attribute__((ext_vector_type(8)));

// D = A(16x4 f32) * B(4x16 f32) + C(16x16 f32), full fp32 precision
#define WMMA_F32(a, b, c) \
  __builtin_amdgcn_wmma_f32_16x16x4_f32(false, (a), false, (b), (short)0, (c), false, false)

// ---------------- degree / norm ----------------
__global__ void deg_init_k(float* __restrict__ deg, int N) {
  int i = blockIdx.x * blockDim.x + threadIdx.x;
  if (i < N) deg[i] = 1.0f;  // self-loop
}

__global__ void deg_count_k(const int* __restrict__ dst, float* __restrict__ deg, int E) {
  int e = blockIdx.x * blockDim.x + threadIdx.x;
  if (e < E) atomicAdd(&deg[dst[e]], 1.0f);
}

__global__ void dinv_k(const float* __restrict__ deg, float* __restrict__ dinv, int N) {
  int i = blockIdx.x * blockDim.x + threadIdx.x;
  if (i < N) dinv[i] = rsqrtf(deg[i]);  // deg >= 1 always
}

// ---------------- layer 1 transform: hs1 = (dinv*x) @ W1; acc1 = hs1 (self-loop seed)
// one wave per 16-node tile; K=6 padded to 8 -> 2 WMMAs per 16-col half (2 halves of 32)
__global__ void __launch_bounds__(256) xform1_k(
    const float* __restrict__ x, const float* __restrict__ W1,
    const float* __restrict__ dinv,
    float* __restrict__ hs1, float* __restrict__ acc1, int N) {
  int wave = (blockIdx.x * blockDim.x + threadIdx.x) >> 5;
  int lane = threadIdx.x & 31;
  int base = wave * 16;
  if (base >= N) return;                 // wave-uniform: EXEC stays all-ones for WMMA
  int half = lane >> 4;                  // lanes 16-31 carry K+2 slice
  int mrow = lane & 15;
  int node = base + mrow; if (node > N - 1) node = N - 1;
  float msel = half ? 0.0f : 1.0f;       // branchless K=6,7 zero-pad

  float dv = dinv[node];                 // pre-scale A rows: (dinv*x)@W1 = dinv*(x@W1)
  const float* xr = x + (long)node * 6;
  v2f a0 = *(const v2f*)(xr + 2 * half); // K = {0,1} | {2,3}   (8B aligned)
  v2f at = *(const v2f*)(xr + 4);        // K = {4,5} raw
  a0.x *= dv;  a0.y *= dv;
  v2f a1;
  a1.x = at.x * (msel * dv);             // K = {4,5} | pad {6,7}=0
  a1.y = at.y * (msel * dv);

  int ka = half ? 2 : 0, kb = half ? 3 : 1;
  v8f c0 = {}, c1 = {};
  {  // output columns 0..15
    v2f b0, bv;
    b0.x = W1[ka * 32 + mrow];
    b0.y = W1[kb * 32 + mrow];
    bv.x = W1[4 * 32 + mrow] * msel;
    bv.y = W1[5 * 32 + mrow] * msel;
    c0 = WMMA_F32(a0, b0, c0);
    c0 = WMMA_F32(a1, bv, c0);
  }
  {  // output columns 16..31
    v2f b0, bv;
    b0.x = W1[ka * 32 + 16 + mrow];
    b0.y = W1[kb * 32 + 16 + mrow];
    bv.x = W1[4 * 32 + 16 + mrow] * msel;
    bv.y = W1[5 * 32 + 16 + mrow] * msel;
    c1 = WMMA_F32(a0, b0, c1);
    c1 = WMMA_F32(a1, bv, c1);
  }

  // C layout: VGPR p -> row p (lanes 0-15) / row p+8 (lanes 16-31); col = lane%16
  if (base + 16 <= N) {                  // full tile: unguarded coalesced stores
#pragma unroll
    for (int p = 0; p < 8; ++p) {
      long o = (long)(base + p + 8 * half) * 32;
      hs1[o + mrow] = c0[p];       acc1[o + mrow] = c0[p];
      hs1[o + 16 + mrow] = c1[p];  acc1[o + 16 + mrow] = c1[p];
    }
  } else {
#pragma unroll
    for (int p = 0; p < 8; ++p) {
      int m = base + p + 8 * half;
      if (m < N) {
        long o = (long)m * 32;
        hs1[o + mrow] = c0[p];       acc1[o + mrow] = c0[p];
        hs1[o + 16 + mrow] = c1[p];  acc1[o + 16 + mrow] = c1[p];
      }
    }
  }
}

// ---------------- layer 1 scatter: acc1[dst] += hs1[src] (32 f32 per edge)
__global__ void scat1_k(const int* __restrict__ src, const int* __restrict__ dst,
                        const float* __restrict__ hs1, float* __restrict__ acc1, int E) {
  int e = blockIdx.x * blockDim.x + threadIdx.x;
  if (e >= E) return;
  long s = src[e], d = dst[e];
  const float4* hp = (const float4*)(hs1 + s * 32);
  float* ap = acc1 + d * 32;
#pragma unroll
  for (int q = 0; q < 8; ++q) {
    float4 v = hp[q];
    atomicAdd(ap + q * 4 + 0, v.x);
    atomicAdd(ap + q * 4 + 1, v.y);
    atomicAdd(ap + q * 4 + 2, v.z);
    atomicAdd(ap + q * 4 + 3, v.w);
  }
}

// ---------------- layer 2 transform, fused finalize of layer 1:
// h = relu(b1 + dinv*acc1); hs2 = (dinv*h) @ W2; acc2 = hs2
__global__ void __launch_bounds__(256) xform2_k(
    const float* __restrict__ acc1, const float* __restrict__ W2,
    const float* __restrict__ bias1, const float* __restrict__ dinv,
    float* __restrict__ hs2, float* __restrict__ acc2, int N) {
  int wave = (blockIdx.x * blockDim.x + threadIdx.x) >> 5;
  int lane = threadIdx.x & 31;
  int base = wave * 16;
  if (base >= N) return;
  int half = lane >> 4;
  int mrow = lane & 15;
  int node = base + mrow; if (node > N - 1) node = N - 1;

  float dv = dinv[node];
  const float* hr = acc1 + (long)node * 32;
  v8f c = {};
#pragma unroll
  for (int s = 0; s < 8; ++s) {          // 8 chained K=4 slices (total K=32)
    int ka = 4 * s + 2 * half;           // even -> 8B aligned float2 reads
    v2f hv = *(const v2f*)(hr + ka);
    v2f bb = *(const v2f*)(bias1 + ka);
    v2f a, b;
    a.x = dv * fmaxf(bb.x + dv * hv.x, 0.0f);   // inner dv: layer-1 norm; outer: layer-2 row pre-scale
    a.y = dv * fmaxf(bb.y + dv * hv.y, 0.0f);
    b.x = W2[ka * 16 + mrow];
    b.y = W2[(ka + 1) * 16 + mrow];
    c = WMMA_F32(a, b, c);
  }

  if (base + 16 <= N) {
#pragma unroll
    for (int p = 0; p < 8; ++p) {
      long o = (long)(base + p + 8 * half) * 16 + mrow;
      hs2[o] = c[p];
      acc2[o] = c[p];
    }
  } else {
#pragma unroll
    for (int p = 0; p < 8; ++p) {
      int m = base + p + 8 * half;
      if (m < N) {
        long o = (long)m * 16 + mrow;
        hs2[o] = c[p];
        acc2[o] = c[p];
      }
    }
  }
}

// ---------------- layer 2 scatter: acc2[dst] += hs2[src] (16 f32 per edge)
__global__ void scat2_k(const int* __restrict__ src, const int* __restrict__ dst,
                        const float* __restrict__ hs2, float* __restrict__ acc2, int E) {
  int e = blockIdx.x * blockDim.x + threadIdx.x;
  if (e >= E) return;
  long s = src[e], d = dst[e];
  const float4* hp = (const float4*)(hs2 + s * 16);
  float* ap = acc2 + d * 16;
#pragma unroll
  for (int q = 0; q < 4; ++q) {
    float4 v = hp[q];
    atomicAdd(ap + q * 4 + 0, v.x);
    atomicAdd(ap + q * 4 + 1, v.y);
    atomicAdd(ap + q * 4 + 2, v.z);
    atomicAdd(ap + q * 4 + 3, v.w);
  }
}

// ---------------- final: z = b2 + dinv * acc2
__global__ void fin2_k(const float* __restrict__ acc2, const float* __restrict__ dinv,
                       const float* __restrict__ b2, float* __restrict__ out, int N) {
  int i = blockIdx.x * blockDim.x + threadIdx.x;
  if (i < N * 16) {
    int node = i >> 4;
    int d = i & 15;
    out[i] = b2[d] + dinv[node] * acc2[i];
  }
}

extern "C" void kernel_launch(void* const* d_in, const int* in_sizes, int n_in,
                              void* d_out, int out_size, void* d_ws, size_t ws_size,
                              hipStream_t stream) {
  const float* x   = (const float*)d_in[0];
  const int*   ei  = (const int*)d_in[1];   // [2, E] flattened
  const float* W1  = (const float*)d_in[2];
  const float* b1  = (const float*)d_in[3];
  const float* W2  = (const float*)d_in[4];
  const float* b2  = (const float*)d_in[5];
  float* out = (float*)d_out;

  const int N = in_sizes[0] / 6;
  const int E = in_sizes[1] / 2;
  const int* src = ei;
  const int* dst = ei + E;

  char* ws = (char*)d_ws;
  size_t off = 0;
  auto alloc = [&](size_t bytes) -> float* {
    float* p = (float*)(ws + off);
    off = (off + bytes + 255) & ~(size_t)255;
    return p;
  };
  float* deg  = alloc((size_t)N * 4);
  float* dinv = alloc((size_t)N * 4);
  float* hs1  = alloc((size_t)N * 32 * 4);
  float* acc1 = alloc((size_t)N * 32 * 4);
  float* hs2  = alloc((size_t)N * 16 * 4);
  float* acc2 = alloc((size_t)N * 16 * 4);

  const int B = 256;              // 8 wave32 per block
  const int tiles = (N + 15) / 16;
  const int tileBlocks = (tiles + 7) / 8;

  deg_init_k<<<(N + B - 1) / B, B, 0, stream>>>(deg, N);
  deg_count_k<<<(E + B - 1) / B, B, 0, stream>>>(dst, deg, E);
  dinv_k<<<(N + B - 1) / B, B, 0, stream>>>(deg, dinv, N);

  xform1_k<<<tileBlocks, B, 0, stream>>>(x, W1, dinv, hs1, acc1, N);
  scat1_k<<<(E + B - 1) / B, B, 0, stream>>>(src, dst, hs1, acc1, E);

  xform2_k<<<tileBlocks, B, 0, stream>>>(acc1, W2, b1, dinv, hs2, acc2, N);
  scat2_k<<<(E + B - 1) / B, B, 0, stream>>>(src, dst, hs2, acc2, E);

  fin2_k<<<(N * 16 + B - 1) / B, B, 0, stream>>>(acc2, dinv, b2, out, N);
}